// FraudDetectionGCN_83648783057204
// MI455X (gfx1250) — compile-verified
//
#include <hip/hip_runtime.h>
#include <hip/hip_bf16.h>

typedef __attribute__((ext_vector_type(2))) float v2f;
typedef __attribute__((ext_vector_type(8))) float v8f;

#define HID 64

// ---------------- degree / normalization ----------------
__global__ void deg_init_kernel(float* deg, int N) {
    int i = blockIdx.x * blockDim.x + threadIdx.x;
    if (i < N) deg[i] = 1.0f;  // self loop
}

__global__ void deg_accum_kernel(const long long* __restrict__ dst, float* deg, int E) {
    int e = blockIdx.x * blockDim.x + threadIdx.x;
    if (e < E) atomicAdd(&deg[dst[e]], 1.0f);
}

__global__ void dinv_kernel(float* deg, int N) {
    int i = blockIdx.x * blockDim.x + threadIdx.x;
    if (i < N) deg[i] = rsqrtf(deg[i]);  // deg >= 1 always (self loops)
}

// ---------------- WMMA fp32 GEMM: [N x K] @ [K x 64] -> [N x 64] ----------------
// One 256-thread block = 8 waves; wave w owns the 16-row tile (blockIdx.x*8 + w).
// W staged in LDS in K-pair-interleaved layout so each lane's v2f B-frag is one
// aligned ds_load_b64:  sW[(k>>1)*128 + c*2 + (k&1)] = W[k*64 + c]
__global__ __launch_bounds__(256) void gemm_h64_wmma_kernel(
    const float* __restrict__ A, const float* __restrict__ W,
    float* __restrict__ out, int N, int K) {
    __shared__ float sW[64 * 128];  // max K=128 -> 64 pairs x 128 floats = 32 KB

    const int tid = threadIdx.x;
    const int total = K * HID;
    for (int i = tid; i < total; i += 256) {
        const int k = i >> 6;          // 0..K-1
        const int c = i & 63;          // column
        sW[(k >> 1) * 128 + c * 2 + (k & 1)] = W[i];
    }
    __syncthreads();

    const int wave = tid >> 5;
    const int lane = tid & 31;
    const int tile = blockIdx.x * 8 + wave;
    if (tile * 16 >= N) return;  // wave-uniform exit; EXEC all-1 for WMMA waves

    const int row0 = tile * 16;
    const int half = lane >> 4;  // 0: K=k0,k0+1 ; 1: K=k0+2,k0+3
    const int l16  = lane & 15;

    const float* __restrict__ Arow = A + (size_t)(row0 + l16) * K;

    v8f acc0 = {}, acc1 = {}, acc2 = {}, acc3 = {};

    for (int k0 = 0; k0 < K; k0 += 4) {
        // A frag: lane (half,l16) holds A[row0+l16, k0+2*half .. +1] -> global_load_b64
        const v2f a = *(const v2f*)(Arow + k0 + 2 * half);

        // B frags: pair index = (k0 + 2*half)/2 = k0/2 + half; contiguous per lane
        const float* pr = sW + (size_t)((k0 >> 1) + half) * 128;
        const v2f b0 = *(const v2f*)(pr + 2 * l16);
        const v2f b1 = *(const v2f*)(pr + 2 * (l16 + 16));
        const v2f b2 = *(const v2f*)(pr + 2 * (l16 + 32));
        const v2f b3 = *(const v2f*)(pr + 2 * (l16 + 48));

        acc0 = __builtin_amdgcn_wmma_f32_16x16x4_f32(false, a, false, b0, (short)0, acc0, false, false);
        acc1 = __builtin_amdgcn_wmma_f32_16x16x4_f32(false, a, false, b1, (short)0, acc1, false, false);
        acc2 = __builtin_amdgcn_wmma_f32_16x16x4_f32(false, a, false, b2, (short)0, acc2, false, false);
        acc3 = __builtin_amdgcn_wmma_f32_16x16x4_f32(false, a, false, b3, (short)0, acc3, false, false);
    }

    // C/D layout: VGPR v, lanes 0-15 -> row M=v, lanes 16-31 -> row M=v+8; col = l16
#pragma unroll
    for (int v = 0; v < 8; ++v) {
        const int rr = row0 + v + 8 * half;
        float* o = out + (size_t)rr * HID + l16;
        o[0]  = acc0[v];
        o[16] = acc1[v];
        o[32] = acc2[v];
        o[48] = acc3[v];
    }
}

// ---------------- aggregation ----------------
// agg[i,f] = h[i,f] * dinv[i]^2  (self-loop term; also initializes agg)
__global__ void selfloop_init_kernel(const float* __restrict__ h, const float* __restrict__ dinv,
                                     float* __restrict__ agg, long long total) {
    long long idx = (long long)blockIdx.x * blockDim.x + threadIdx.x;
    if (idx < total) {
        long long node = idx >> 6;
        float d = dinv[node];
        agg[idx] = h[idx] * d * d;
    }
}

// one WAVE per edge; each lane owns 2 features (float2 row read, 2 f32 atomics).
// src/dst/dinv reads are wave-uniform (one broadcast cacheline per edge).
__global__ __launch_bounds__(256) void edge_scatter_kernel(
    const long long* __restrict__ src, const long long* __restrict__ dst,
    const float* __restrict__ h, const float* __restrict__ dinv,
    float* __restrict__ agg, long long E) {
    const long long gid = (long long)blockIdx.x * blockDim.x + threadIdx.x;
    const long long e = gid >> 5;           // wave id == edge id
    const int lane = threadIdx.x & 31;
    if (e >= E) return;

    const long long s = src[e];
    const long long d = dst[e];
    const float w = dinv[s] * dinv[d];

    const int f = lane * 2;
    const v2f hv = *(const v2f*)(h + s * HID + f);
    float* o = agg + d * HID + f;
    atomicAdd(o + 0, hv.x * w);
    atomicAdd(o + 1, hv.y * w);
}

__global__ void bias_relu_kernel(float* __restrict__ agg, const float* __restrict__ b, long long total) {
    long long idx = (long long)blockIdx.x * blockDim.x + threadIdx.x;
    if (idx < total) {
        float v = agg[idx] + b[idx & 63];
        agg[idx] = v > 0.0f ? v : 0.0f;
    }
}

// ---------------- final FC (64 -> 2) + log_softmax ----------------
__global__ void fc_logsoftmax_kernel(const float* __restrict__ h, const float* __restrict__ Wfc,
                                     const float* __restrict__ bfc, float* __restrict__ out, int N) {
    int i = blockIdx.x * blockDim.x + threadIdx.x;
    if (i >= N) return;
    const float* hr = h + (size_t)i * HID;
    float l0 = bfc[0], l1 = bfc[1];
#pragma unroll 8
    for (int f = 0; f < HID; ++f) {
        float v = hr[f];
        l0 = fmaf(v, Wfc[f * 2 + 0], l0);
        l1 = fmaf(v, Wfc[f * 2 + 1], l1);
    }
    float m = fmaxf(l0, l1);
    float lse = m + logf(expf(l0 - m) + expf(l1 - m));
    out[(size_t)i * 2 + 0] = l0 - lse;
    out[(size_t)i * 2 + 1] = l1 - lse;
}

// ---------------- launcher ----------------
extern "C" void kernel_launch(void* const* d_in, const int* in_sizes, int n_in,
                              void* d_out, int out_size, void* d_ws, size_t ws_size,
                              hipStream_t stream) {
    const float*     x   = (const float*)d_in[0];
    const long long* ei  = (const long long*)d_in[1];
    const float*     W1  = (const float*)d_in[2];
    const float*     b1  = (const float*)d_in[3];
    const float*     W2  = (const float*)d_in[4];
    const float*     b2  = (const float*)d_in[5];
    const float*     W3  = (const float*)d_in[6];
    const float*     b3  = (const float*)d_in[7];
    const float*     Wfc = (const float*)d_in[8];
    const float*     bfc = (const float*)d_in[9];
    float* out = (float*)d_out;

    const int D = 128;
    const int N = in_sizes[0] / D;          // 100000
    const int E = in_sizes[1] / 2;          // 1600000
    const long long NH = (long long)N * HID;

    const long long* src = ei;
    const long long* dst = ei + E;

    // workspace: dinv [N] | tmp [N*64] | agg [N*64]
    float* dinv = (float*)d_ws;
    float* tmp  = dinv + N;
    float* agg  = tmp + NH;

    const int TPB = 256;
    const int gN    = (N + TPB - 1) / TPB;
    const int gE    = (E + TPB - 1) / TPB;
    const int gNH   = (int)((NH + TPB - 1) / TPB);
    const int gScat = (int)(((long long)E * 32 + TPB - 1) / TPB);  // 1 wave per edge
    const int gGemm = (((N + 15) / 16) + 7) / 8;                   // 8 row-tiles per block

    // normalization
    deg_init_kernel<<<gN, TPB, 0, stream>>>(dinv, N);
    deg_accum_kernel<<<gE, TPB, 0, stream>>>(dst, dinv, E);
    dinv_kernel<<<gN, TPB, 0, stream>>>(dinv, N);

    // layer 1: x[N,128] @ W1[128,64]
    gemm_h64_wmma_kernel<<<gGemm, TPB, 0, stream>>>(x, W1, tmp, N, D);
    selfloop_init_kernel<<<gNH, TPB, 0, stream>>>(tmp, dinv, agg, NH);
    edge_scatter_kernel<<<gScat, TPB, 0, stream>>>(src, dst, tmp, dinv, agg, E);
    bias_relu_kernel<<<gNH, TPB, 0, stream>>>(agg, b1, NH);

    // layer 2
    gemm_h64_wmma_kernel<<<gGemm, TPB, 0, stream>>>(agg, W2, tmp, N, HID);
    selfloop_init_kernel<<<gNH, TPB, 0, stream>>>(tmp, dinv, agg, NH);
    edge_scatter_kernel<<<gScat, TPB, 0, stream>>>(src, dst, tmp, dinv, agg, E);
    bias_relu_kernel<<<gNH, TPB, 0, stream>>>(agg, b2, NH);

    // layer 3
    gemm_h64_wmma_kernel<<<gGemm, TPB, 0, stream>>>(agg, W3, tmp, N, HID);
    selfloop_init_kernel<<<gNH, TPB, 0, stream>>>(tmp, dinv, agg, NH);
    edge_scatter_kernel<<<gScat, TPB, 0, stream>>>(src, dst, tmp, dinv, agg, E);
    bias_relu_kernel<<<gNH, TPB, 0, stream>>>(agg, b3, NH);

    // classifier + log_softmax
    fc_logsoftmax_kernel<<<gN, TPB, 0, stream>>>(agg, Wfc, bfc, out, N);
}